// SIA_36704790511691
// MI455X (gfx1250) — compile-verified
//
#include <hip/hip_runtime.h>
#include <hip/hip_bf16.h>
#include <math.h>

// ---------------------------------------------------------------------------
// DGCNN + SuperGlue-attention pipeline for MI455X (gfx1250), wave32.
// All GEMMs use exact-f32 V_WMMA_F32_16X16X4_F32 (A 16x4, B 4x16, C 16x16).
// GEMMs templated on channel count: fully unrolled K-loop, no guarded loads,
// EXEC always all-ones. Optional fused BN+ReLU on the B operand (BNIN) kills
// the 25MB elementwise passes. Flash attention: 2 q-tiles/wave (2x k/v reuse),
// software-pipelined k prefetch, DPP16 butterfly reductions (no ds_bpermute
// in the hot loop), per-lane online-softmax partial sums, hardware __expf.
// Deterministic (no float atomics).
// ---------------------------------------------------------------------------

typedef float v2f __attribute__((ext_vector_type(2)));
typedef float v8f __attribute__((ext_vector_type(8)));

#define WMMA_F32(a, b, c) \
    __builtin_amdgcn_wmma_f32_16x16x4_f32(false, (a), false, (b), (short)0, (c), false, false)

namespace sia {

constexpr int   B    = 4;
constexpr int   N    = 2048;
constexpr int   KNNK = 12;
constexpr int   D    = 64;
constexpr int   H    = 4;       // heads, dh = 16
constexpr float EPS  = 1e-5f;

constexpr int FX   = B * D * N;            // 524288
constexpr int FBIG = B * D * N * KNNK;     // 6291456

// -------------------- DPP16 butterfly reductions (VALU-only) --------------
// Patterns stay inside a 16-lane DPP row == our half-wave reduction groups.
// 0xB1 quad_perm[1,0,3,2] (xor1), 0x4E quad_perm[2,3,0,1] (xor2),
// 0x141 row_half_mirror (8-group), 0x140 row_mirror (16-group).

template <int CTRL>
__device__ __forceinline__ float dpp_movc(float x) {
    int xi = __float_as_int(x);
    return __int_as_float(__builtin_amdgcn_update_dpp(xi, xi, CTRL, 0xf, 0xf, true));
}
__device__ __forceinline__ float red16_max(float x) {
    x = fmaxf(x, dpp_movc<0xB1>(x));
    x = fmaxf(x, dpp_movc<0x4E>(x));
    x = fmaxf(x, dpp_movc<0x141>(x));
    x = fmaxf(x, dpp_movc<0x140>(x));
    return x;
}
__device__ __forceinline__ float red16_sum(float x) {
    x += dpp_movc<0xB1>(x);
    x += dpp_movc<0x4E>(x);
    x += dpp_movc<0x141>(x);
    x += dpp_movc<0x140>(x);
    return x;
}

// -------------------- utility kernels --------------------

__global__ void k_zero(float* __restrict__ p, int n) {
    int i = blockIdx.x * blockDim.x + threadIdx.x;
    if (i < n) p[i] = 0.f;
}

__global__ void k_add2(float* __restrict__ o, const float* __restrict__ a,
                       const float* __restrict__ b, int n) {
    int i = blockIdx.x * blockDim.x + threadIdx.x;
    if (i < n) o[i] = a[i] + b[i];
}

__global__ void k_add3(float* __restrict__ o, const float* __restrict__ a,
                       const float* __restrict__ b, const float* __restrict__ c, int n) {
    int i = blockIdx.x * blockDim.x + threadIdx.x;
    if (i < n) o[i] = a[i] + b[i] + c[i];
}

// per-batch strided copy: dst[b*dstB + r] = src[b*srcB + r], r < perB
__global__ void k_copy_rows(const float* __restrict__ src, float* __restrict__ dst,
                            int perB, int srcB, int dstB) {
    int i = blockIdx.x * blockDim.x + threadIdx.x;
    if (i >= perB * B) return;
    int b = i / perB, r = i - b * perB;
    dst[b * dstB + r] = src[b * srcB + r];
}

// -------------------- kNN (top-12 by squared distance, incl. self) ---------

__global__ void k_knn(const float* __restrict__ x, int* __restrict__ idx) {
    __shared__ float sx[3 * N];                 // 24 KB
    int b = blockIdx.y;
    const float* xb = x + b * 3 * N;
    for (int i = threadIdx.x; i < 3 * N; i += 256) sx[i] = xb[i];
    __syncthreads();

    int n = blockIdx.x * 256 + threadIdx.x;
    float px = sx[n], py = sx[N + n], pz = sx[2 * N + n];

    float bd[KNNK]; int bi[KNNK];
#pragma unroll
    for (int p = 0; p < KNNK; ++p) { bd[p] = -INFINITY; bi[p] = 0; }

    for (int m = 0; m < N; ++m) {
        float dx = px - sx[m], dy = py - sx[N + m], dz = pz - sx[2 * N + m];
        float d = -(dx * dx + dy * dy + dz * dz);   // neg squared distance
        if (d > bd[KNNK - 1]) {
            bd[KNNK - 1] = d; bi[KNNK - 1] = m;
#pragma unroll
            for (int p = KNNK - 1; p > 0; --p) {
                if (bd[p] > bd[p - 1]) {
                    float td = bd[p]; bd[p] = bd[p - 1]; bd[p - 1] = td;
                    int   ti = bi[p]; bi[p] = bi[p - 1]; bi[p - 1] = ti;
                }
            }
        }
    }
    int* o = idx + (b * N + n) * KNNK;
#pragma unroll
    for (int p = 0; p < KNNK; ++p) o[p] = bi[p];
}

// -------------------- generic f32 WMMA GEMM (C % 4 == 0, templated) --------
// Y[b][m][s] = sum_c W[m][c] * op(X[b][c][s])  (+ bias[m])  (+ Y_old if ACC)
// op = identity, or fused BN+ReLU (BNIN): relu(x*scl[c] + shf[c]).
// One wave computes a 16x16 tile; fully unrolled K-loop, no guards.

template <int C, bool BIAS, bool ACC, bool BNIN>
__global__ void k_gemm(const float* __restrict__ W, const float* __restrict__ bias,
                       const float* __restrict__ X, float* Y,
                       int S, int xB, int yB,
                       const float* __restrict__ scl, const float* __restrict__ shf) {
    int lane = threadIdx.x, half = lane >> 4, l16 = lane & 15;
    int s0 = blockIdx.x * 16;
    int m0 = blockIdx.y * 16;
    int b  = blockIdx.z;
    const float* Xb = X + b * xB + s0 + l16;
    float*       Yb = Y + b * yB;

    v8f acc = {};
#pragma unroll
    for (int v = 0; v < 8; ++v) {
        int row = m0 + v + 8 * half;
        float init = 0.f;
        if (BIAS) init += bias[row];
        if (ACC)  init += Yb[row * S + s0 + l16];
        acc[v] = init;
    }
    const v2f* Wp = (const v2f*)(W + (m0 + l16) * C);   // 8B-aligned: C even
#pragma unroll
    for (int k = 0; k < C; k += 4) {
        int ka = k + 2 * half;
        v2f a = Wp[(k >> 1) + half];
        float bx = Xb[ka * S];
        float by = Xb[(ka + 1) * S];
        if (BNIN) {
            v2f sc2 = *(const v2f*)(scl + ka);          // ka even: 8B-aligned
            v2f sh2 = *(const v2f*)(shf + ka);
            bx = fmaxf(fmaf(bx, sc2.x, sh2.x), 0.f);
            by = fmaxf(fmaf(by, sc2.y, sh2.y), 0.f);
        }
        v2f bf; bf.x = bx; bf.y = by;
        acc = WMMA_F32(a, bf, acc);
    }
#pragma unroll
    for (int v = 0; v < 8; ++v) {
        int row = m0 + v + 8 * half;
        Yb[row * S + s0 + l16] = acc[v];
    }
}

// EdgeConv GEMM: B-matrix built on the fly from neighbor gather:
// d[c][n*12+kk] = x[c][idx[n][kk]] - x[c][n].  CIN is 3 or 64.
template <int CIN>
__global__ void k_gemm_edge(const float* __restrict__ W, const float* __restrict__ X,
                            const int* __restrict__ idx, float* __restrict__ Y,
                            int xB, int yB) {
    int lane = threadIdx.x, half = lane >> 4, l16 = lane & 15;
    const int S = N * KNNK;
    int s0 = blockIdx.x * 16;
    int m0 = blockIdx.y * 16;
    int b  = blockIdx.z;
    const float* Xb = X + b * xB;
    const int*   ib = idx + b * N * KNNK;
    float*       Yb = Y + b * yB;

    int col = s0 + l16;
    int n  = col / KNNK;
    int kk = col - n * KNNK;
    int j  = ib[n * KNNK + kk];

    v8f acc = {};
    if (CIN == 3) {
        // Single K-step, zero-pad lane-half 1's .y via value select (no branches).
        int row = m0 + l16;
        int ka  = 2 * half;                  // 0 or 2 (always valid)
        int kys = half ? 2 : 1;              // safe row for the .y element
        v2f a, bf;
        a.x = W[row * 3 + ka];
        float ay = W[row * 3 + kys];
        a.y = half ? 0.f : ay;
        bf.x = Xb[ka * N + j] - Xb[ka * N + n];
        float by = Xb[kys * N + j] - Xb[kys * N + n];
        bf.y = half ? 0.f : by;
        acc = WMMA_F32(a, bf, acc);
    } else {
        const v2f* Wp = (const v2f*)(W + (m0 + l16) * CIN);
#pragma unroll
        for (int k = 0; k < CIN; k += 4) {
            int ka = k + 2 * half;
            v2f a = Wp[(k >> 1) + half];
            v2f bf;
            bf.x = Xb[ka * N + j]       - Xb[ka * N + n];
            bf.y = Xb[(ka + 1) * N + j] - Xb[(ka + 1) * N + n];
            acc = WMMA_F32(a, bf, acc);
        }
    }
#pragma unroll
    for (int v = 0; v < 8; ++v) {
        int row = m0 + v + 8 * half;
        Yb[row * S + s0 + l16] = acc[v];
    }
}

// -------------------- training-mode BatchNorm --------------------
// Stage 1: per-channel mean/var over (B, S); emits scale = g*rsqrt(var+eps),
// shift = beta - mean*scale. One block per channel, deterministic tree reduce.
__global__ void k_bn_stats(const float* __restrict__ X, const float* __restrict__ g,
                           const float* __restrict__ bet, float* __restrict__ scale,
                           float* __restrict__ shift, int S, int bstride) {
    __shared__ float rs[256], rq[256];
    int c = blockIdx.x, tid = threadIdx.x;
    float s = 0.f, q = 0.f;
    for (int b = 0; b < B; ++b) {
        const float* p = X + b * bstride + c * S;
        for (int i = tid; i < S; i += 256) {
            float v = p[i];
            s += v; q += v * v;
        }
    }
    rs[tid] = s; rq[tid] = q; __syncthreads();
    for (int o = 128; o > 0; o >>= 1) {
        if (tid < o) { rs[tid] += rs[tid + o]; rq[tid] += rq[tid + o]; }
        __syncthreads();
    }
    if (tid == 0) {
        float inv  = 1.f / (float)(B * S);
        float mean = rs[0] * inv;
        float var  = rq[0] * inv - mean * mean;
        float sc   = g[c] * rsqrtf(var + EPS);
        scale[c] = sc;
        shift[c] = bet[c] - mean * sc;
    }
}

// Stage 2 (standalone): y = relu(x*scale[c] + shift[c])
__global__ void k_bn_relu(const float* __restrict__ X, float* __restrict__ Y,
                          const float* __restrict__ scale, const float* __restrict__ shift,
                          int C, int S, int total) {
    int i = blockIdx.x * blockDim.x + threadIdx.x;
    if (i >= total) return;
    int c = (i / S) % C;
    float v = fmaf(X[i], scale[c], shift[c]);
    Y[i] = v > 0.f ? v : 0.f;
}

// Fused BN + ReLU + max over K neighbors (relu∘max == max∘relu).
__global__ void k_bn_relu_maxk(const float* __restrict__ X, float* __restrict__ Y,
                               const float* __restrict__ scale, const float* __restrict__ shift) {
    int g = blockIdx.x * blockDim.x + threadIdx.x;
    if (g >= FX) return;
    int c = (g / N) % D;                 // g enumerates (b, c, n)
    float sc = scale[c], sh = shift[c];
    const float* p = X + g * KNNK;       // 12 contiguous pre-BN values
    float m = -INFINITY;
#pragma unroll
    for (int kk = 0; kk < KNNK; ++kk) m = fmaxf(m, fmaf(p[kk], sc, sh));
    Y[g] = fmaxf(m, 0.f);
}

// -------------------- flash attention --------------------
// One wave handles TWO 16-row q-tiles for one (b, head): k/v fragments are
// loaded once per m-tile and feed both tiles. q/k/v: [B][64][N]; head h uses
// channels c = i*H + h (dh = 16). 1/sqrt(dh) = 0.25 folded into q (exact).

__device__ __forceinline__ void sm_update(v8f& sc, float (&rmax)[8], float (&rsum)[8],
                                          v8f& o, float* pl, int half, int l16) {
#pragma unroll
    for (int vv = 0; vv < 8; ++vv) {
        float s  = sc[vv];
        float tm = red16_max(s);                 // DPP16 butterfly, no LDS
        float nm    = fmaxf(rmax[vv], tm);
        float alpha = __expf(rmax[vv] - nm);     // row-uniform across the group
        float p     = __expf(s - nm);
        rsum[vv] = rsum[vv] * alpha + p;         // per-lane partial sum
        rmax[vv] = nm;
        o[vv] *= alpha;
        pl[(vv + 8 * half) * 16 + l16] = p;      // P row-major [n][m]
    }
}

__device__ __forceinline__ void sm_store(v8f& o, float (&rsum)[8], float* op) {
    float r[8];
#pragma unroll
    for (int vv = 0; vv < 8; ++vv) r[vv] = o[vv] / red16_sum(rsum[vv]);
    float4 lo, hi;
    lo.x = r[0]; lo.y = r[1]; lo.z = r[2]; lo.w = r[3];
    hi.x = r[4]; hi.y = r[5]; hi.z = r[6]; hi.w = r[7];
    *(float4*)(op)     = lo;
    *(float4*)(op + 4) = hi;
}

__global__ void k_attn_flash(const float* __restrict__ q, const float* __restrict__ k,
                             const float* __restrict__ v, float* __restrict__ msg) {
    __shared__ float plds[512];
    int lane = threadIdx.x, half = lane >> 4, l16 = lane & 15;
    int n0 = blockIdx.x * 32, h = blockIdx.y, b = blockIdx.z;
    const float* qb = q + b * D * N;
    const float* kb = k + b * D * N;
    const float* vb = v + b * D * N;
    float*       mb = msg + b * D * N;

    // Preload scaled q A-fragments for both tiles (invariant over m-tiles).
    v2f af0[4], af1[4];
#pragma unroll
    for (int t = 0; t < 4; ++t) {
        int ka = 4 * t + 2 * half;
        const float* r0 = qb + (ka * H + h) * N;
        const float* r1 = qb + ((ka + 1) * H + h) * N;
        af0[t].x = 0.25f * r0[n0 + l16];
        af0[t].y = 0.25f * r1[n0 + l16];
        af1[t].x = 0.25f * r0[n0 + 16 + l16];
        af1[t].y = 0.25f * r1[n0 + 16 + l16];
    }
    const float* vrow = vb + (l16 * H + h) * N;       // fixed v row per lane

    float rmax0[8], rsum0[8], rmax1[8], rsum1[8];
    v8f o0 = {}, o1 = {};
#pragma unroll
    for (int vv = 0; vv < 8; ++vv) {
        rmax0[vv] = -INFINITY; rsum0[vv] = 0.f;
        rmax1[vv] = -INFINITY; rsum1[vv] = 0.f;
    }

    // Software pipeline: prefetch k fragments for tile 0.
    v2f kf[4];
#pragma unroll
    for (int t = 0; t < 4; ++t) {
        int ka = 4 * t + 2 * half;
        kf[t].x = kb[(ka * H + h) * N + l16];
        kf[t].y = kb[((ka + 1) * H + h) * N + l16];
    }

    for (int mt = 0; mt < N / 16; ++mt) {
        int m0 = mt * 16;
        v2f kcur[4];
#pragma unroll
        for (int t = 0; t < 4; ++t) kcur[t] = kf[t];

        // Prefetch next m-tile's k fragments (wrapped index: always valid,
        // no branch). Loads overlap the softmax VALU work below.
        int mnext = ((mt + 1) & (N / 16 - 1)) * 16;
#pragma unroll
        for (int t = 0; t < 4; ++t) {
            int ka = 4 * t + 2 * half;
            kf[t].x = kb[(ka * H + h) * N + mnext + l16];
            kf[t].y = kb[((ka + 1) * H + h) * N + mnext + l16];
        }
        // Prefetch this tile's v fragments (consumed after softmax).
        const v2f* vrt = (const v2f*)(vrow + m0);
        v2f bv[4];
#pragma unroll
        for (int t = 0; t < 4; ++t) bv[t] = vrt[2 * t + half];

        // ---- S tiles = qT * k (2x 16n x 16m), shared k fragments ----
        v8f sc0 = {}, sc1 = {};
#pragma unroll
        for (int t = 0; t < 4; ++t) {
            sc0 = WMMA_F32(af0[t], kcur[t], sc0);
            sc1 = WMMA_F32(af1[t], kcur[t], sc1);
        }
        // ---- online softmax (DPP16 max, per-lane partial sums) ----
        sm_update(sc0, rmax0, rsum0, o0, plds,       half, l16);
        sm_update(sc1, rmax1, rsum1, o1, plds + 256, half, l16);
        __syncthreads();
        // ---- O += P * vT (2 tiles, shared v fragments) ----
        const v2f* p0 = (const v2f*)(plds + l16 * 16);
        const v2f* p1 = (const v2f*)(plds + 256 + l16 * 16);
#pragma unroll
        for (int t = 0; t < 4; ++t) {
            int kp = 2 * t + half;
            o0 = WMMA_F32(p0[kp], bv[t], o0);
            o1 = WMMA_F32(p1[kp], bv[t], o1);
        }
        __syncthreads();
    }
    // finalize: one DPP sum reduction per row, then 16B stores
    float* opbase = mb + (l16 * H + h) * N;
    sm_store(o0, rsum0, opbase + n0 + 8 * half);
    sm_store(o1, rsum1, opbase + n0 + 16 + 8 * half);
}

// -------------------- final: out = x3 * (1 + exp(softmax_row(emb))) --------
__global__ void k_final(const float* __restrict__ emb, const float* __restrict__ x3,
                        float* __restrict__ out) {
    __shared__ float red[256];
    int tid = threadIdx.x;
    int base = blockIdx.x * N;
    const float* e = emb + base;
    const float* x = x3 + base;

    float m = -INFINITY;
    for (int i = tid; i < N; i += 256) m = fmaxf(m, e[i]);
    red[tid] = m; __syncthreads();
    for (int o = 128; o > 0; o >>= 1) { if (tid < o) red[tid] = fmaxf(red[tid], red[tid + o]); __syncthreads(); }
    m = red[0]; __syncthreads();

    float s = 0.f;
    for (int i = tid; i < N; i += 256) s += expf(e[i] - m);
    red[tid] = s; __syncthreads();
    for (int o = 128; o > 0; o >>= 1) { if (tid < o) red[tid] += red[tid + o]; __syncthreads(); }
    s = red[0]; __syncthreads();

    float inv = 1.f / s;
    for (int i = tid; i < N; i += 256) {
        float p = expf(e[i] - m) * inv;
        out[base + i] = x[i] * (1.f + expf(p));
    }
}

} // namespace sia

// ---------------------------------------------------------------------------

extern "C" void kernel_launch(void* const* d_in, const int* in_sizes, int n_in,
                              void* d_out, int out_size, void* d_ws, size_t ws_size,
                              hipStream_t stream) {
    using namespace sia;
    (void)in_sizes; (void)n_in; (void)out_size; (void)ws_size;

    auto F = [&](int i) { return (const float*)d_in[i]; };
    const float* x = F(0);
    // params in setup_inputs() insertion order:
    // p1..p3: w2a,g2a,b2a,w2b,g2b,b2b,w1,g1,b1  (9 each, starting at 1)
    // a1..a3: wq,bq,wk,bk,wv,bv,wm,bm,w1,c1,g1,e1,w2,c2 (14 each, starting at 28)
    const float* P1[9]; const float* P2[9]; const float* P3[9];
    for (int i = 0; i < 9; ++i) { P1[i] = F(1 + i); P2[i] = F(10 + i); P3[i] = F(19 + i); }
    const float* A1[14]; const float* A2[14]; const float* A3[14];
    for (int i = 0; i < 14; ++i) { A1[i] = F(28 + i); A2[i] = F(42 + i); A3[i] = F(56 + i); }

    // Workspace layout (floats). Requires ~78.5 MB of d_ws.
    float* w    = (float*)d_ws;
    float* bigA = w;                 // [B][64][N*12]
    float* bigB = bigA + FBIG;       // [B][64][N*12]
    float* x1   = bigB + FBIG;
    float* x2   = x1 + FX;
    float* x3   = x2 + FX;
    float* tmp  = x3 + FX;
    float* qb   = tmp + FX;
    float* kb   = qb + FX;
    float* vb   = kb + FX;
    float* msgb = vb + FX;
    float* embb = msgb + FX;
    float* h2   = embb + FX;                    // [B][128][N]
    float* h1   = h2 + B * 128 * N;             // [B][128][N]
    float* scl  = h1 + B * 128 * N;             // 128
    float* shf  = scl + 128;                    // 128
    int*   idxb = (int*)(shf + 128);            // [B][N][12]

    const int SNK   = N * KNNK;                 // 24576
    const int bigBS = D * SNK;                  // per-batch stride, big buffers
    const int xBS   = D * N;                    // per-batch stride, [64][N]
    const int hBS   = 128 * N;

    dim3 blk256(256), blk32(32);
    auto gcd1 = [](int n) { return dim3((unsigned)((n + 255) / 256)); };

    // ---- zero emb accumulator (deterministic across graph replays) ----
    k_zero<<<gcd1(FX), blk256, 0, stream>>>(embb, FX);

    // ---- kNN ----
    k_knn<<<dim3(N / 256, B), blk256, 0, stream>>>(x, idxb);

    dim3 gEdge(SNK / 16, D / 16, B);
    dim3 gX64(N / 16, D / 16, B);
    dim3 gX128(N / 16, 128 / 16, B);

    // ---- EdgeConv propagate stage ----
    auto propagate = [&](const float* xin, const float* const* P, bool first, float* dst) {
        // conv2a on gathered diffs (WMMA) -> bigA (raw, pre-BN)
        if (first)
            k_gemm_edge<3><<<gEdge, blk32, 0, stream>>>(P[0], xin, idxb, bigA, 3 * N, bigBS);
        else
            k_gemm_edge<64><<<gEdge, blk32, 0, stream>>>(P[0], xin, idxb, bigA, xBS, bigBS);
        k_bn_stats<<<dim3(D), blk256, 0, stream>>>(bigA, P[1], P[2], scl, shf, SNK, bigBS);
        // conv2b with fused BN+ReLU on input (WMMA) -> bigB (raw, pre-BN)
        k_gemm<64, false, false, true><<<gEdge, blk32, 0, stream>>>(
            P[3], nullptr, bigA, bigB, SNK, bigBS, bigBS, scl, shf);
        k_bn_stats<<<dim3(D), blk256, 0, stream>>>(bigB, P[4], P[5], scl, shf, SNK, bigBS);
        // fused BN + ReLU + max over K -> qb (scratch)
        k_bn_relu_maxk<<<gcd1(FX), blk256, 0, stream>>>(bigB, qb, scl, shf);
        // conv1 (WMMA) -> kb (scratch), BN+ReLU -> dst
        k_gemm<64, false, false, false><<<gX64, blk32, 0, stream>>>(
            P[6], nullptr, qb, kb, N, xBS, xBS, nullptr, nullptr);
        k_bn_stats<<<dim3(D), blk256, 0, stream>>>(kb, P[7], P[8], scl, shf, N, xBS);
        k_bn_relu<<<gcd1(FX), blk256, 0, stream>>>(kb, dst, scl, shf, D, N, FX);
    };

    // ---- attention stage (accumulates into embb) ----
    auto attention = [&](const float* xin, const float* const* A) {
        k_gemm<64, true, false, false><<<gX64, blk32, 0, stream>>>(
            A[0], A[1], xin, qb, N, xBS, xBS, nullptr, nullptr);
        k_gemm<64, true, false, false><<<gX64, blk32, 0, stream>>>(
            A[2], A[3], xin, kb, N, xBS, xBS, nullptr, nullptr);
        k_gemm<64, true, false, false><<<gX64, blk32, 0, stream>>>(
            A[4], A[5], xin, vb, N, xBS, xBS, nullptr, nullptr);
        k_attn_flash<<<dim3(N / 32, H, B), blk32, 0, stream>>>(qb, kb, vb, msgb);
        // wm*msg + bm -> rows 64..127 of h2 ; copy xin -> rows 0..63
        k_gemm<64, true, false, false><<<gX64, blk32, 0, stream>>>(
            A[6], A[7], msgb, h2 + D * N, N, xBS, hBS, nullptr, nullptr);
        k_copy_rows<<<gcd1(FX), blk256, 0, stream>>>(xin, h2, xBS, xBS, hBS);
        // MLP: w1 (128x128) + c1 -> h1 (raw); BN stats; w2 with fused BN+ReLU
        // on input, + c2, accumulating into embb.
        k_gemm<128, true, false, false><<<gX128, blk32, 0, stream>>>(
            A[8], A[9], h2, h1, N, hBS, hBS, nullptr, nullptr);
        k_bn_stats<<<dim3(128), blk256, 0, stream>>>(h1, A[10], A[11], scl, shf, N, hBS);
        k_gemm<128, true, true, true><<<gX64, blk32, 0, stream>>>(
            A[12], A[13], h1, embb, N, hBS, xBS, scl, shf);
    };

    // ---- pipeline ----
    propagate(x, P1, true, x1);
    propagate(x1, P2, false, tmp);
    k_add2<<<gcd1(FX), blk256, 0, stream>>>(x2, x1, tmp, FX);
    propagate(x2, P3, false, tmp);
    k_add3<<<gcd1(FX), blk256, 0, stream>>>(x3, x1, x2, tmp, FX);

    attention(x1, A1);
    attention(x2, A2);
    attention(x3, A3);

    k_final<<<dim3(B * D), blk256, 0, stream>>>(embb, x3, (float*)d_out);
}